// RicciTensorNetwork_74732430950676
// MI455X (gfx1250) — compile-verified
//
#include <hip/hip_runtime.h>
#include <hip/hip_bf16.h>

// ---------------------------------------------------------------------------
// RicciTensorNetwork on gfx1250 (CDNA5), fp32 end-to-end via V_WMMA_F32_16X16X4_F32.
// Sizes: N=16, H=128, B=256.
//   metric:      points(256x16) -> relu(@mW1 16x128) -> @mW2 128x256 -> sym + eps*I
//   christoffel: 1M triples, [m_ij,m_jk,m_ki] -> tanh(@cW1 3x128) -> @cW2 128x1
//   ricci:       65536 rows (b,i,j): 64 -> relu(@rW1) -> 256; the trailing GEMM2,
//                symmetrization and (i,j)-sum are linear, so we sum hidden first:
//                S[b] = (sum_ij relu(...)) @ rW2 + 256*rb2 ; ricci = 0.5(S+S^T)/256
// ---------------------------------------------------------------------------

typedef float v2f __attribute__((ext_vector_type(2)));
typedef float v8f __attribute__((ext_vector_type(8)));

#define NDIM 16
#define HDIM 128
#define BATCH 256

__device__ __forceinline__ v8f wmma_f32_16x16x4(v2f a, v2f b, v8f c) {
  // 8 args: (neg_a, A, neg_b, B, c_mod, C, reuse_a, reuse_b)
  return __builtin_amdgcn_wmma_f32_16x16x4_f32(false, a, false, b, (short)0, c,
                                               false, false);
}

__device__ __forceinline__ float fast_tanh(float x) {
  // tanh(x) = 1 - 2/(e^{2x}+1), uses v_exp_f32 + v_rcp path
  float e = __expf(2.0f * x);
  return 1.0f - 2.0f / (e + 1.0f);
}

// ---------------------------------------------------------------------------
// Kernel A: metric.  One workgroup (256 thr = 8 waves) per 16 batch rows.
// Fragment conventions (16x16x4 f32, wave32):
//   A: lanes 0-15 row M=l16 hold {K=k0,K=k0+1} with k0=kb; lanes 16-31 k0=kb+2.
//   B: lanes 0-15 col N=l16 hold rows {kb,kb+1}; lanes 16-31 rows {kb+2,kb+3}.
//   D: VGPR r -> (M=r, N=l16) for lanes 0-15, (M=r+8, N=l16) for lanes 16-31.
// ---------------------------------------------------------------------------
__global__ void __launch_bounds__(256)
metric_kernel(const float* __restrict__ points,
              const float* __restrict__ mW1, const float* __restrict__ mb1,
              const float* __restrict__ mW2, const float* __restrict__ mb2,
              float* __restrict__ metric) {
  __shared__ float sP[16 * 16];        // points tile
  __shared__ float sW1[16 * 128];      // layer-1 weights
  __shared__ float sMh[16 * 128];      // relu hidden
  __shared__ float sRaw[16 * 256];     // raw metric rows
  const int tid  = threadIdx.x;
  const int lane = tid & 31;
  const int wave = tid >> 5;
  const int half = lane >> 4;
  const int l16  = lane & 15;
  const int rowbase = blockIdx.x * 16;

  sP[tid] = points[rowbase * NDIM + tid];
  for (int t = tid; t < 16 * 128; t += 256) sW1[t] = mW1[t];
  __syncthreads();

  // -------- layer 1: hidden col-tile ct = wave (8 waves x 16 cols = 128) ----
  {
    const int ct = wave;
    v8f acc = {};
    for (int kk = 0; kk < 4; ++kk) {
      const int k0 = 4 * kk + half * 2;
      v2f a, b;
      a.x = sP[l16 * 16 + k0];
      a.y = sP[l16 * 16 + k0 + 1];
      b.x = sW1[k0 * 128 + ct * 16 + l16];
      b.y = sW1[(k0 + 1) * 128 + ct * 16 + l16];
      acc = wmma_f32_16x16x4(a, b, acc);
    }
    const float bias = mb1[ct * 16 + l16];
    for (int r = 0; r < 8; ++r) {
      float v = acc[r] + bias;
      v = v > 0.0f ? v : 0.0f;
      sMh[(r + half * 8) * 128 + ct * 16 + l16] = v;
    }
  }
  __syncthreads();

  // -------- layer 2: output col-tiles {wave, wave+8} (16 x 16 = 256) -------
  for (int tt = 0; tt < 2; ++tt) {
    const int ct = wave + tt * 8;
    v8f acc = {};
    for (int kk = 0; kk < 32; ++kk) {
      const int k0 = 4 * kk + half * 2;
      v2f a, b;
      a.x = sMh[l16 * 128 + k0];
      a.y = sMh[l16 * 128 + k0 + 1];
      b.x = mW2[k0 * 256 + ct * 16 + l16];          // mW2 stays in L2
      b.y = mW2[(k0 + 1) * 256 + ct * 16 + l16];
      acc = wmma_f32_16x16x4(a, b, acc);
    }
    const float bias = mb2[ct * 16 + l16];
    for (int r = 0; r < 8; ++r)
      sRaw[(r + half * 8) * 256 + ct * 16 + l16] = acc[r] + bias;
  }
  __syncthreads();

  // -------- symmetrize + eps*I ---------------------------------------------
  for (int t = tid; t < 16 * 256; t += 256) {
    const int row = t >> 8, col = t & 255;
    const int i = col >> 4, j = col & 15;
    float v = 0.5f * (sRaw[row * 256 + i * 16 + j] +
                      sRaw[row * 256 + j * 16 + i]);
    if (i == j) v += 1e-6f;
    metric[(rowbase + row) * 256 + col] = v;
  }
}

// ---------------------------------------------------------------------------
// Kernel B: christoffel.  One wave per (b,i,j) tile of 16 triples (k=0..15).
// Layer-1 is a single K=4 WMMA per 16-hidden tile with input row
// [m_ij, m_jk, m_ki, 1.0] and cb1 folded in as weight row K=3.
// ---------------------------------------------------------------------------
__global__ void __launch_bounds__(256)
christoffel_kernel(const float* __restrict__ metric,
                   const float* __restrict__ cW1, const float* __restrict__ cb1,
                   const float* __restrict__ cW2, const float* __restrict__ cb2,
                   float* __restrict__ chris) {
  const int tid  = threadIdx.x;
  const int lane = tid & 31;
  const int wave = tid >> 5;
  const int half = lane >> 4;
  const int l16  = lane & 15;
  const int tile = blockIdx.x * 8 + wave;   // flat (b,i,j), 65536 tiles
  const int b = tile >> 8;
  const int i = (tile >> 4) & 15;
  const int j = tile & 15;
  const float* m = metric + b * 256;

  // Loop-invariant weight fragments + layer-2 weights: load once per wave.
  v2f bfrag[8];
  float w2[8];
  for (int ct = 0; ct < 8; ++ct) {
    if (half == 0) {
      bfrag[ct].x = cW1[0 * 128 + ct * 16 + l16];  // row K=0
      bfrag[ct].y = cW1[1 * 128 + ct * 16 + l16];  // row K=1
    } else {
      bfrag[ct].x = cW1[2 * 128 + ct * 16 + l16];  // row K=2
      bfrag[ct].y = cb1[ct * 16 + l16];            // row K=3 = bias
    }
    w2[ct] = cW2[ct * 16 + l16];
  }

  // A fragment: rows = the 16 k-values of this (b,i,j)
  v2f a;
  if (half == 0) {
    a.x = m[i * 16 + j];      // K=0: m_ij  (broadcast down rows)
    a.y = m[j * 16 + l16];    // K=1: m_jk  (row = k = l16)
  } else {
    a.x = m[l16 * 16 + i];    // K=2: m_ki
    a.y = 1.0f;               // K=3: bias lane
  }

  float partial[8];
  for (int r = 0; r < 8; ++r) partial[r] = 0.0f;

  for (int ct = 0; ct < 8; ++ct) {          // 8 x 16 = 128 hidden
    v8f c = {};
    v8f d = wmma_f32_16x16x4(a, bfrag[ct], c);
    for (int r = 0; r < 8; ++r)
      partial[r] += fast_tanh(d[r]) * w2[ct];
  }

  // cross-lane reduce over the 16 lanes holding the 16 hidden columns
  for (int mask = 1; mask <= 8; mask <<= 1)
    for (int r = 0; r < 8; ++r)
      partial[r] += __shfl_xor(partial[r], mask, 32);

  if (l16 == 0) {
    const float c2 = cb2[0];
    for (int r = 0; r < 8; ++r)
      chris[tile * 16 + half * 8 + r] = partial[r] + c2;  // rows M=r(+8) = k
  }
}

// ---------------------------------------------------------------------------
// Kernel C: ricci.  One workgroup per batch element b.
// GEMM2 + symmetrize + sum are linear => sum hidden (after relu) over all
// (i,j), then one 256x256 mat-vec.  GEMM1 via WMMA over 16 row-tiles (i=rt).
// rW1 B-fragments are invariant across the 16 row-tiles -> preload to VGPRs.
// ---------------------------------------------------------------------------
__global__ void __launch_bounds__(256)
ricci_kernel(const float* __restrict__ points,
             const float* __restrict__ metric,
             const float* __restrict__ chris,
             const float* __restrict__ rW1, const float* __restrict__ rb1,
             const float* __restrict__ rW2, const float* __restrict__ rb2,
             float* __restrict__ out) {
  __shared__ float sIn[16 * 64];    // r_in tile: rows = j, cols = 64 features
  __shared__ float sHsum[256];      // sum_{i,j} relu hidden
  __shared__ float sT[256];         // S[b] row
  const int tid  = threadIdx.x;
  const int lane = tid & 31;
  const int wave = tid >> 5;
  const int half = lane >> 4;
  const int l16  = lane & 15;
  const int b = blockIdx.x;

  sHsum[tid] = 0.0f;

  // Preload rW1 B-fragments for this wave's two hidden col-tiles
  // (constant across all 16 row-tiles): 2 x 16 x v2f = 64 VGPRs/lane.
  v2f bfrag[2][16];
  for (int tt = 0; tt < 2; ++tt) {
    const int ct = wave * 2 + tt;
    for (int kk = 0; kk < 16; ++kk) {
      const int k0 = 4 * kk + half * 2;
      bfrag[tt][kk].x = rW1[k0 * 256 + ct * 16 + l16];
      bfrag[tt][kk].y = rW1[(k0 + 1) * 256 + ct * 16 + l16];
    }
  }

  // Warm L2/L0 for the mat-vec weights used at the end.
  __builtin_prefetch(&rW2[(tid & 255) * 256], 0, 1);
  __syncthreads();

  for (int rt = 0; rt < 16; ++rt) {         // i = rt, rows m = j
    for (int t = tid; t < 16 * 64; t += 256) {
      const int mrow = t >> 6, c = t & 63;
      float v;
      if (c < 16)      v = points[b * 16 + c];
      else if (c < 32) v = metric[b * 256 + rt * 16 + (c - 16)];     // metric[i,:]
      else if (c < 48) v = metric[b * 256 + mrow * 16 + (c - 32)];   // metric[j,:]
      else             v = chris[b * 4096 + rt * 256 + mrow * 16 + (c - 48)];
      sIn[mrow * 64 + c] = v;
    }
    __syncthreads();

    for (int tt = 0; tt < 2; ++tt) {        // 8 waves x 2 = 16 hidden tiles
      const int ct = wave * 2 + tt;
      v8f acc = {};
      for (int kk = 0; kk < 16; ++kk) {     // K = 64, pure LDS + WMMA
        const int k0 = 4 * kk + half * 2;
        v2f a;
        a.x = sIn[l16 * 64 + k0];
        a.y = sIn[l16 * 64 + k0 + 1];
        acc = wmma_f32_16x16x4(a, bfrag[tt][kk], acc);
      }
      const float bias = rb1[ct * 16 + l16];
      float colsum = 0.0f;
      for (int r = 0; r < 8; ++r) {
        float v = acc[r] + bias;
        colsum += (v > 0.0f ? v : 0.0f);
      }
      atomicAdd(&sHsum[ct * 16 + l16], colsum);      // ds_add_f32
    }
    __syncthreads();
  }

  // collapsed GEMM2: S = Hsum @ rW2 + 256*rb2   (one column per thread)
  float t = 256.0f * rb2[tid];
  for (int h = 0; h < 256; ++h)
    t += sHsum[h] * rW2[h * 256 + tid];
  sT[tid] = t;
  __syncthreads();

  const int k = tid >> 4, l = tid & 15;
  out[b * 256 + tid] = 0.5f * (sT[tid] + sT[l * 16 + k]) * (1.0f / 256.0f);
}

// ---------------------------------------------------------------------------
extern "C" void kernel_launch(void* const* d_in, const int* in_sizes, int n_in,
                              void* d_out, int out_size, void* d_ws, size_t ws_size,
                              hipStream_t stream) {
  const float* points = (const float*)d_in[0];
  const float* mW1 = (const float*)d_in[1];
  const float* mb1 = (const float*)d_in[2];
  const float* mW2 = (const float*)d_in[3];
  const float* mb2 = (const float*)d_in[4];
  const float* cW1 = (const float*)d_in[5];
  const float* cb1 = (const float*)d_in[6];
  const float* cW2 = (const float*)d_in[7];
  const float* cb2 = (const float*)d_in[8];
  const float* rW1 = (const float*)d_in[9];
  const float* rb1 = (const float*)d_in[10];
  const float* rW2 = (const float*)d_in[11];
  const float* rb2 = (const float*)d_in[12];

  float* metric_ws = (float*)d_ws;                       // 256*256 f32 = 256KB
  float* chris_ws  = metric_ws + BATCH * NDIM * NDIM;    // 256*4096 f32 = 4MB
  float* outf      = (float*)d_out;

  metric_kernel<<<BATCH / 16, 256, 0, stream>>>(points, mW1, mb1, mW2, mb2,
                                                metric_ws);
  christoffel_kernel<<<(BATCH * NDIM * NDIM) / 8, 256, 0, stream>>>(
      metric_ws, cW1, cb1, cW2, cb2, chris_ws);
  ricci_kernel<<<BATCH, 256, 0, stream>>>(points, metric_ws, chris_ws,
                                          rW1, rb1, rW2, rb2, outf);
}